// GRNEncoder_24567212933619
// MI455X (gfx1250) — compile-verified
//
#include <hip/hip_runtime.h>
#include <math.h>

// Problem constants (from reference)
#define NB   8
#define NT   32
#define NN   500
#define NF   32
#define NHID 64
#define NGH  512
#define NE   8000
#define DIN  (NN * NHID)   // 32000
#define G3   (3 * NGH)     // 1536
#define MROWS (NB * NT)    // 256

typedef float v2f __attribute__((ext_vector_type(2)));
typedef float v8f __attribute__((ext_vector_type(8)));
typedef int   v2i __attribute__((ext_vector_type(2)));

#if __has_builtin(__builtin_amdgcn_global_load_async_to_lds_b64)
#define HAVE_ASYNC_LDS 1
typedef __attribute__((address_space(1))) v2i* g_v2i_p;
typedef __attribute__((address_space(3))) v2i* l_v2i_p;
#else
#define HAVE_ASYNC_LDS 0
#endif

__device__ __forceinline__ void wait_async_lds() {
#if HAVE_ASYNC_LDS
#if __has_builtin(__builtin_amdgcn_s_wait_asynccnt)
    __builtin_amdgcn_s_wait_asynccnt(0);
#else
    asm volatile("s_wait_asynccnt 0" ::: "memory");
#endif
#endif
}

// ---------------- small utility kernels ----------------

__global__ void fill_kernel(float* p, float v, int n) {
    int i = blockIdx.x * blockDim.x + threadIdx.x;
    if (i < n) p[i] = v;
}

// deg starts at 1.0 (self-loop); add 1 per incoming edge (col side)
__global__ void deg_scatter_kernel(const long long* ei, float* deg) {
    int e = blockIdx.x * blockDim.x + threadIdx.x;
    if (e < NE) {
        int c = (int)ei[NE + e];       // col = edge_index[1]
        atomicAdd(&deg[c], 1.0f);
    }
}

__global__ void dinv_kernel(const float* deg, float* dinv) {
    int n = blockIdx.x * blockDim.x + threadIdx.x;
    if (n < NN) dinv[n] = rsqrtf(deg[n]);   // deg >= 1 always
}

__global__ void norm_kernel(const long long* ei, const float* dinv, float* norm) {
    int e = blockIdx.x * blockDim.x + threadIdx.x;
    if (e < NE) {
        int r = (int)ei[e];
        int c = (int)ei[NE + e];
        norm[e] = dinv[r] * dinv[c];
    }
}

// xagg init = self-loop contribution: dinv[n]^2 * x
__global__ void xagg_init_kernel(const float* x, const float* dinv, float* xagg) {
    int i = blockIdx.x * blockDim.x + threadIdx.x;   // over B*T*N*F = 4,096,000
    if (i < MROWS * NN * NF) {
        int n = (i / NF) % NN;
        float d = dinv[n];
        xagg[i] = d * d * x[i];
    }
}

// scatter messages in F-space: one wave per (bt, edge), 32 lanes = features
__global__ void edge_scatter_kernel(const float* x, const long long* ei,
                                    const float* norm, float* xagg) {
    long long idx = (long long)blockIdx.x * blockDim.x + threadIdx.x;
    if (idx >= (long long)MROWS * NE * NF) return;
    int f  = (int)(idx & 31);
    int e  = (int)((idx >> 5) % NE);
    int bt = (int)(idx / ((long long)NE * NF));
    int r = (int)ei[e];
    int c = (int)ei[NE + e];
    float v = norm[e] * x[(bt * NN + r) * NF + f];
    atomicAdd(&xagg[(bt * NN + c) * NF + f], v);
}

// node_emb = relu(xagg @ W + b); write as gcn_flat (256 x 32000)
__global__ void gcn_linear_relu_kernel(const float* xagg, const float* W,
                                       const float* bias, float* gflat) {
    int blk = blockIdx.x;           // bt*NN + n
    int bt = blk / NN;
    int n  = blk % NN;
    int h  = threadIdx.x;           // 0..63
    __shared__ float sx[NF];
    if (h < NF) sx[h] = xagg[(bt * NN + n) * NF + h];
    __syncthreads();
    float acc = bias[h];
    #pragma unroll
    for (int f = 0; f < NF; ++f)
        acc = fmaf(sx[f], W[f * NHID + h], acc);
    gflat[bt * DIN + n * NHID + h] = fmaxf(acc, 0.0f);
}

__global__ void gi_init_kernel(float* GI, const float* bih) {
    int i = blockIdx.x * blockDim.x + threadIdx.x;   // over 256*1536
    if (i < MROWS * G3) GI[i] = bih[i % G3];
}

// ---------------- big GEMM: GI += gcn_flat(256x32000) @ w_ih^T ----------------
// grid = (16 n-blocks of 96 cols, 20 k-splits), block = 512 threads (16 waves).
// A (256 x 64-k chunk) is double-buffered in LDS via async global->LDS loads:
// the stage for chunk ch+1 is issued before computing chunk ch and only waited
// (s_wait_asynccnt 0 + barrier) afterwards. B (w_ih) streams global->VGPR with
// a one-step register double-buffer so loads for k-step s+1 overlap the six
// WMMAs of step s. One HBM pass over the 197MB w_ih is the roofline floor.
#define NACC 6                    // n-tiles per wave -> block covers 96 cols
#define NPB  (NACC * 16)          // 96
#define KSPLIT 20
#define CH_PER_SPLIT 25           // 25 * 64 * 20 = 32000
#define ASTRIDE 68                // padded LDS row stride (floats)

__global__ void __launch_bounds__(512)
big_gemm_kernel(const float* __restrict__ A, const float* __restrict__ Wih,
                float* __restrict__ GI) {
    __shared__ float as_[2][256 * ASTRIDE];     // 2 x 69.6 KB

    const int lane = threadIdx.x & 31;
    const int wv   = threadIdx.x >> 5;          // 0..15 -> M tile
    const int m0   = wv * 16;
    const int n0   = blockIdx.x * NPB;
    const int kb0  = blockIdx.y * CH_PER_SPLIT * 64;

    v8f acc[NACC];
    #pragma unroll
    for (int a = 0; a < NACC; ++a) acc[a] = (v8f){};

    const int arow = m0 + (lane & 15);
    const int koff = (lane < 16) ? 0 : 2;       // lanes 0-15: K0,K1; 16-31: K2,K3
    const int nn   = lane & 15;
    const int k2   = lane * 2;                  // staging k offset per lane

    // per-tile base pointers for the B stream
    const float* bp[NACC];
    #pragma unroll
    for (int t = 0; t < NACC; ++t)
        bp[t] = Wih + (size_t)(n0 + t * 16 + nn) * DIN + koff;

    // stage A chunk `kb` into LDS buffer `buf` (async if available)
    auto stage = [&](int buf, int kb) {
        #pragma unroll 4
        for (int r = 0; r < 16; ++r) {
            const int m = r * 16 + wv;
            const size_t gidx = (size_t)m * DIN + kb + k2;
            const int    lidx = m * ASTRIDE + k2;
#if HAVE_ASYNC_LDS
            __builtin_amdgcn_global_load_async_to_lds_b64(
                (g_v2i_p)&A[gidx], (l_v2i_p)&as_[buf][lidx], 0, 0);
#else
            v2f w = *(const v2f*)&A[gidx];
            *(v2f*)&as_[buf][lidx] = w;
#endif
        }
    };

    stage(0, kb0);
    wait_async_lds();
    __syncthreads();

    for (int ch = 0; ch < CH_PER_SPLIT; ++ch) {
        const int kb  = kb0 + ch * 64;
        const int buf = ch & 1;
        // issue next chunk's stage into the other buffer; wait only after compute
        if (ch + 1 < CH_PER_SPLIT) stage(buf ^ 1, kb + 64);

        // preload B fragments for k-step 0 of this chunk
        v2f bcur[NACC];
        #pragma unroll
        for (int t = 0; t < NACC; ++t) bcur[t] = *(const v2f*)&bp[t][kb];

        #pragma unroll
        for (int s = 0; s < 16; ++s) {
            v2f bnxt[NACC];
            if (s < 15) {
                #pragma unroll
                for (int t = 0; t < NACC; ++t)
                    bnxt[t] = *(const v2f*)&bp[t][kb + (s + 1) * 4];
            }
            v2f a = *(const v2f*)&as_[buf][arow * ASTRIDE + s * 4 + koff];
            #pragma unroll
            for (int t = 0; t < NACC; ++t)
                acc[t] = __builtin_amdgcn_wmma_f32_16x16x4_f32(
                    false, a, false, bcur[t], (short)0, acc[t], false, false);
            if (s < 15) {
                #pragma unroll
                for (int t = 0; t < NACC; ++t) bcur[t] = bnxt[t];
            }
        }

        wait_async_lds();      // next chunk's async stage complete (this wave)
        __syncthreads();       // visible to all waves; prev buffer reusable
    }

    // C layout: VGPR r, lanes 0-15 -> M=r; lanes 16-31 -> M=r+8; N = lane%16
    const int Mbase = (lane >> 4) << 3;
    #pragma unroll
    for (int t = 0; t < NACC; ++t) {
        #pragma unroll
        for (int r = 0; r < 8; ++r) {
            int row = m0 + r + Mbase;
            atomicAdd(&GI[row * G3 + n0 + t * 16 + nn], acc[t][r]);
        }
    }
}

// ---------------- GRU step (one kernel per timestep) ----------------
// grid = 32 blocks (16-col gate tiles), block = 32 threads (1 wave).
// h buffers are 16 x 512 with rows 8..15 kept zero (WMMA M padding).
// A/B fragments are register double-buffered to hide the serial load chain.
__global__ void gru_step_kernel(const float* __restrict__ GI,
                                const float* __restrict__ Whh,
                                const float* __restrict__ bhh,
                                const float* __restrict__ hin,
                                float* __restrict__ hout, int t) {
    const int lane = threadIdx.x & 31;
    const int j0   = blockIdx.x * 16;
    const int mrow = lane & 15;
    const int koff = (lane < 16) ? 0 : 2;
    const int nn   = lane & 15;

    const float* ap  = hin + (size_t)mrow * NGH + koff;
    const float* bpr = Whh + (size_t)(j0 + nn) * NGH + koff;
    const float* bpz = Whh + (size_t)(NGH + j0 + nn) * NGH + koff;
    const float* bpn = Whh + (size_t)(2 * NGH + j0 + nn) * NGH + koff;

    v8f ar = {}, az = {}, an = {};
    v2f ac  = *(const v2f*)&ap[0];
    v2f brc = *(const v2f*)&bpr[0];
    v2f bzc = *(const v2f*)&bpz[0];
    v2f bnc = *(const v2f*)&bpn[0];
    #pragma unroll 8
    for (int k4 = 0; k4 < NGH; k4 += 4) {
        v2f anx, brx, bzx, bnx;
        if (k4 + 4 < NGH) {
            anx = *(const v2f*)&ap[k4 + 4];
            brx = *(const v2f*)&bpr[k4 + 4];
            bzx = *(const v2f*)&bpz[k4 + 4];
            bnx = *(const v2f*)&bpn[k4 + 4];
        }
        ar = __builtin_amdgcn_wmma_f32_16x16x4_f32(false, ac, false, brc, (short)0, ar, false, false);
        az = __builtin_amdgcn_wmma_f32_16x16x4_f32(false, ac, false, bzc, (short)0, az, false, false);
        an = __builtin_amdgcn_wmma_f32_16x16x4_f32(false, ac, false, bnc, (short)0, an, false, false);
        if (k4 + 4 < NGH) { ac = anx; brc = brx; bzc = bzx; bnc = bnx; }
    }
    if (lane < 16) {   // lanes 0-15 hold M = r (batch rows 0..7 are in r=0..7)
        const int col = j0 + lane;
        const float bh_r = bhh[col];
        const float bh_z = bhh[NGH + col];
        const float bh_n = bhh[2 * NGH + col];
        #pragma unroll
        for (int r = 0; r < 8; ++r) {
            const int b = r;                       // batch index
            const int gi_row = b * NT + t;         // gcn_flat flattening b*T + t
            float gir = GI[gi_row * G3 + col];
            float giz = GI[gi_row * G3 + NGH + col];
            float gin = GI[gi_row * G3 + 2 * NGH + col];
            float hr = ar[r] + bh_r;
            float hz = az[r] + bh_z;
            float hn = an[r] + bh_n;
            float rg = 1.0f / (1.0f + expf(-(gir + hr)));
            float zg = 1.0f / (1.0f + expf(-(giz + hz)));
            float ng = tanhf(gin + rg * hn);
            float ho = hin[b * NGH + col];
            hout[b * NGH + col] = (1.0f - zg) * ng + zg * ho;
        }
    }
}

__global__ void copy_out_kernel(float* out, const float* h) {
    int i = blockIdx.x * blockDim.x + threadIdx.x;
    if (i < NB * NGH) out[i] = h[i];   // rows 0..7 contiguous
}

// ---------------- launch ----------------

extern "C" void kernel_launch(void* const* d_in, const int* in_sizes, int n_in,
                              void* d_out, int out_size, void* d_ws, size_t ws_size,
                              hipStream_t stream) {
    const float*     x   = (const float*)d_in[0];
    const long long* ei  = (const long long*)d_in[1];   // int64 (2, E)
    const float*     gw  = (const float*)d_in[2];
    const float*     gb  = (const float*)d_in[3];
    const float*     wih = (const float*)d_in[4];
    const float*     whh = (const float*)d_in[5];
    const float*     bih = (const float*)d_in[6];
    const float*     bhh = (const float*)d_in[7];
    float* out = (float*)d_out;

    // workspace carve-up (floats); total ~50.9 MB
    float* ws    = (float*)d_ws;
    float* deg   = ws;                       // 512
    float* dinv  = ws + 512;                 // 512
    float* norm  = ws + 1024;                // 8192
    float* xagg  = ws + 16384;               // 4,096,000
    float* gflat = xagg + (size_t)MROWS * NN * NF;    // 8,192,000
    float* GI    = gflat + (size_t)MROWS * DIN;       // 393,216
    float* hA    = GI + (size_t)MROWS * G3;           // 8,192
    float* hB    = hA + 16 * NGH;                     // 8,192

    // init: deg = 1 (self-loops), h ping-pong buffers = 0 (hA,hB contiguous)
    fill_kernel<<<2, 256, 0, stream>>>(deg, 1.0f, NN);
    fill_kernel<<<64, 256, 0, stream>>>(hA, 0.0f, 2 * 16 * NGH);
    gi_init_kernel<<<(MROWS * G3 + 255) / 256, 256, 0, stream>>>(GI, bih);

    // GCN normalization
    deg_scatter_kernel<<<(NE + 255) / 256, 256, 0, stream>>>(ei, deg);
    dinv_kernel<<<2, 256, 0, stream>>>(deg, dinv);
    norm_kernel<<<(NE + 255) / 256, 256, 0, stream>>>(ei, dinv, norm);

    // aggregate in F-space, then linear+relu
    xagg_init_kernel<<<(MROWS * NN * NF + 255) / 256, 256, 0, stream>>>(x, dinv, xagg);
    {
        long long tot = (long long)MROWS * NE * NF;   // 65,536,000
        edge_scatter_kernel<<<(unsigned)((tot + 255) / 256), 256, 0, stream>>>(x, ei, norm, xagg);
    }
    gcn_linear_relu_kernel<<<MROWS * NN, NHID, 0, stream>>>(xagg, gw, gb, gflat);

    // batched input-gate GEMM: one pass over the 197MB w_ih (HBM-bound floor)
    big_gemm_kernel<<<dim3(G3 / NPB, KSPLIT), 512, 0, stream>>>(gflat, wih, GI);

    // sequential GRU: 32 small WMMA kernels, ping-pong h
    for (int t = 0; t < NT; ++t) {
        const float* hin = (t & 1) ? hB : hA;
        float*       ho  = (t & 1) ? hA : hB;
        gru_step_kernel<<<NGH / 16, 32, 0, stream>>>(GI, whh, bhh, hin, ho, t);
    }
    // after t=31 (odd), result is in hA
    copy_out_kernel<<<(NB * NGH + 255) / 256, 256, 0, stream>>>(out, hA);
}